// PositionalEncodingLayer_68771016343791
// MI455X (gfx1250) — compile-verified
//
#include <hip/hip_runtime.h>
#include <stdint.h>

#define D_EMBED 1024
#define SEQ     4096
#define BATCH   4
#define SCALE   32.0f   // sqrt(1024)

// -(2/1024) * log2(10000)
#define NEG_EXP_STEP (-0.025952563241307517f)
#define INV_2PI      (0.15915494309189535f)

typedef float v4f __attribute__((ext_vector_type(4)));

__global__ __launch_bounds__(256)
void pe_add_kernel(const float* __restrict__ emb, float* __restrict__ out) {
    // Staging buffer: one row (1024 cols) per batch, 16 KB total.
    __shared__ float lds_buf[BATCH * D_EMBED];

    const int tid = threadIdx.x;
    const int s   = blockIdx.x;       // sequence position, 0..4095
    const int c0  = tid * 4;          // first of this thread's 4 columns
    const size_t row = (size_t)s * D_EMBED + c0;

    // ---- Phase 1: issue async global->LDS loads for all 4 batches (ASYNCcnt) ----
#pragma unroll
    for (int b = 0; b < BATCH; ++b) {
        const float* gp = emb + (size_t)b * (size_t)SEQ * D_EMBED + row;
        // Low 32 bits of a flat pointer into LDS == LDS byte offset.
        uint32_t lds_off = (uint32_t)(uintptr_t)(&lds_buf[b * D_EMBED + c0]);
        asm volatile("global_load_async_to_lds_b128 %0, %1, off"
                     :: "v"(lds_off), "v"(gp)
                     : "memory");
    }

    // ---- Phase 2: compute pe for this thread's 4 columns while loads fly ----
    const float pos = (float)s;
    float pe[4];
#pragma unroll
    for (int j = 0; j < 4; ++j) {
        const float fi  = (float)(c0 + j);
        const float inv = __builtin_amdgcn_exp2f(fi * NEG_EXP_STEP); // 10000^(-2i/D)
        const float ang = pos * inv;                                  // radians
        // Range-reduce to [0,1) turns; hw v_sin/v_cos take input in revolutions.
        const float r = ang * INV_2PI;
        const float f = r - floorf(r);
        pe[j] = (j & 1) ? __builtin_amdgcn_cosf(f) : __builtin_amdgcn_sinf(f);
    }

    // ---- Phase 3: wait for async copies, then fused scale+add and stream out ----
    asm volatile("s_wait_asynccnt 0" ::: "memory");

#pragma unroll
    for (int b = 0; b < BATCH; ++b) {
        const v4f v = *(const v4f*)(&lds_buf[b * D_EMBED + c0]);
        v4f o;
        o.x = fmaf(v.x, SCALE, pe[0]);
        o.y = fmaf(v.y, SCALE, pe[1]);
        o.z = fmaf(v.z, SCALE, pe[2]);
        o.w = fmaf(v.w, SCALE, pe[3]);
        v4f* op = (v4f*)(out + (size_t)b * (size_t)SEQ * D_EMBED + row);
        __builtin_nontemporal_store(o, op);   // write-once stream; keep L2 for reads
    }
}

extern "C" void kernel_launch(void* const* d_in, const int* in_sizes, int n_in,
                              void* d_out, int out_size, void* d_ws, size_t ws_size,
                              hipStream_t stream) {
    (void)in_sizes; (void)n_in; (void)out_size; (void)d_ws; (void)ws_size;
    const float* emb = (const float*)d_in[0];
    float* out = (float*)d_out;
    pe_add_kernel<<<SEQ, 256, 0, stream>>>(emb, out);
}